// LengthRegulator_20332375179470
// MI455X (gfx1250) — compile-verified
//
#include <hip/hip_runtime.h>
#include <stdint.h>

// Problem constants (match reference)
#define B_N        32
#define T_N        256
#define D_N        512
#define MAX_FRAMES 2816                 // T * (DUR_MAX - 1)
#define TILE_ROWS  16                   // frames per TDM tile (gather-mode max)
#define TILES_PB   (MAX_FRAMES / TILE_ROWS)   // 176 tiles per batch
#define VPF        (D_N / 4)            // float4 vectors per frame = 128

typedef unsigned int uint32;
typedef __attribute__((ext_vector_type(4))) float        v4f;
typedef __attribute__((ext_vector_type(4))) unsigned int v4u;
typedef __attribute__((ext_vector_type(8))) int          v8i;
typedef __attribute__((ext_vector_type(4))) int          v4i;

// 32-bit wave-relative LDS byte address (generic shared pointers keep the LDS
// offset in the low 32 bits).
__device__ __forceinline__ uint32 lds_addr_of(const void* p) {
    return (uint32)(uintptr_t)p;
}

// ---------------------------------------------------------------------------
// Tensor DMA Descriptor (D#) builders, per CDNA5 ISA ch.8 (§8.3-8.6).
// ---------------------------------------------------------------------------
__device__ __forceinline__ v4u tdm_group0(uint32 lds_addr, uint64_t gaddr, bool gather) {
    v4u g;
    g[0] = 1u | (gather ? 0x80000000u : 0u);       // count=1; gather_mode bit31;
                                                   // gather_index_size bit30=0 (16-bit)
    g[1] = lds_addr;                               // lds_addr [63:32]
    g[2] = (uint32)gaddr;                          // global_addr [95:64]
    g[3] = ((uint32)(gaddr >> 32) & 0x01FFFFFFu)   // global_addr [120:96]
           | (2u << 30);                           // type=2 ("image") [127:126]
    return g;
}

__device__ __forceinline__ v8i tdm_group1(uint32 dim0, uint32 dim1,
                                          uint32 tile0, uint32 tile1, uint32 tile2,
                                          uint64_t stride0) {
    v8i g;
    g[0] = (int)(2u << 16);                               // wg_mask=0; data_size=2 (4B)
    g[1] = (int)((dim0 & 0xFFFFu) << 16);                 // atomic_bar=0 | tensor_dim0 lo16
    g[2] = (int)((dim0 >> 16) | ((dim1 & 0xFFFFu) << 16));// tensor_dim0 hi16 | tensor_dim1 lo16
    g[3] = (int)((dim1 >> 16) | (tile0 << 16));           // tensor_dim1 hi16 | tile_dim0
    g[4] = (int)((tile1 & 0xFFFFu) | (tile2 << 16));      // tile_dim1 | tile_dim2
    g[5] = (int)(uint32)stride0;                          // tensor_dim0_stride lo32
    g[6] = (int)(uint32)(stride0 >> 32);                  // stride0 hi16 | dim1_stride lo16 (0)
    g[7] = 0;                                             // dim1_stride hi (0)
    return g;
}

// ---------------------------------------------------------------------------
// Kernel 1: per-batch inclusive cumsum of durations (256 tokens / batch).
// ---------------------------------------------------------------------------
__global__ __launch_bounds__(256)
void lr_cumsum_kernel(const int* __restrict__ ds, int* __restrict__ cs) {
    __shared__ int s[T_N];
    const int b = blockIdx.x;
    const int t = threadIdx.x;
    s[t] = ds[b * T_N + t];
    __syncthreads();
#pragma unroll
    for (int off = 1; off < T_N; off <<= 1) {
        int v = (t >= off) ? s[t - off] : 0;
        __syncthreads();
        s[t] += v;
        __syncthreads();
    }
    cs[b * T_N + t] = s[t];
}

// ---------------------------------------------------------------------------
// Kernel 2: one wave per 16-frame tile.
//   nv==0  : pure pad tile -> direct non-temporal zero stores (no LDS/TDM)
//   nv>0   : TDM gather-load of the nv token rows into LDS (one instruction),
//            zero remaining pad rows in LDS, TDM store of the 32KB tile.
// ---------------------------------------------------------------------------
__global__ __launch_bounds__(32)
void lr_tdm_kernel(const float* __restrict__ xs,
                   const int* __restrict__ cs,
                   float* __restrict__ out) {
    __shared__ v4f s_buf[TILE_ROWS * VPF];   // 32 KB tile staging

    const int b    = blockIdx.y;
    const int f0   = blockIdx.x * TILE_ROWS;
    const int lane = threadIdx.x;

    const int* __restrict__ csb = cs + b * T_N;
    const int total = csb[T_N - 1];

    int nv = total - f0;                      // valid frames in this tile
    nv = nv < 0 ? 0 : (nv > TILE_ROWS ? TILE_ROWS : nv);

    v4f* out4 = (v4f*)out;
    const size_t out_base = ((size_t)b * MAX_FRAMES + (size_t)f0) * (size_t)VPF;

    if (nv == 0) {
        // Entire tile is padding: stream zeros, bypass caches.
        const v4f z = (v4f){0.f, 0.f, 0.f, 0.f};
#pragma unroll 4
        for (int i = lane; i < TILE_ROWS * VPF; i += 32)
            __builtin_nontemporal_store(z, &out4[out_base + (size_t)i]);
        return;
    }

    // searchsorted(cs, f, 'right'): lanes 0..15 each resolve one frame's token.
    int tok;
    {
        const int f = f0 + (lane & 15);
        int lo = 0, hi = T_N;
        while (lo < hi) {
            const int mid = (lo + hi) >> 1;
            if (csb[mid] > f) hi = mid; else lo = mid + 1;
        }
        tok = (lo < T_N - 1) ? lo : (T_N - 1);
    }

    // Scalarize the 16 (non-decreasing) row indices into D# groups 2/3
    // (16-bit packed, 2 per dword).
#define RL16(k) ((uint32)__builtin_amdgcn_readlane(tok, (k)) & 0xFFFFu)
    v4i g2, g3;
    g2[0] = (int)(RL16(0)  | (RL16(1)  << 16));
    g2[1] = (int)(RL16(2)  | (RL16(3)  << 16));
    g2[2] = (int)(RL16(4)  | (RL16(5)  << 16));
    g2[3] = (int)(RL16(6)  | (RL16(7)  << 16));
    g3[0] = (int)(RL16(8)  | (RL16(9)  << 16));
    g3[1] = (int)(RL16(10) | (RL16(11) << 16));
    g3[2] = (int)(RL16(12) | (RL16(13) << 16));
    g3[3] = (int)(RL16(14) | (RL16(15) << 16));
#undef RL16

    // Zero the pad rows of the LDS tile (disjoint from the gathered rows).
    const v4f z = (v4f){0.f, 0.f, 0.f, 0.f};
    for (int i = nv * VPF + lane; i < TILE_ROWS * VPF; i += 32)
        s_buf[i] = z;

    // Trailing 8-dword group (VADDR4 slot — unused for <=5D tensors, keep 0).
    const v8i g4z = (v8i){0, 0, 0, 0, 0, 0, 0, 0};

    // TDM gather-load: nv rows of 512 f32, row r from xs[b, tok[r], :].
    const uint64_t xs_base = (uint64_t)(uintptr_t)(xs + (size_t)b * T_N * D_N);
    const v4u lg0 = tdm_group0(lds_addr_of(&s_buf[0]), xs_base, /*gather=*/true);
    const v8i lg1 = tdm_group1(/*dim0*/D_N, /*dim1*/T_N,
                               /*tile0*/D_N, /*tile1*/(uint32)nv, /*tile2*/0,
                               /*stride0*/D_N);
    __builtin_amdgcn_tensor_load_to_lds(lg0, lg1, g2, g3, g4z, /*cpol*/0);

    // Gathered rows in LDS + our DS zero-stores complete before the store issues.
    __builtin_amdgcn_s_wait_tensorcnt(0);
    asm volatile("s_wait_dscnt 0" ::: "memory");

    // TDM store: contiguous 16x512 f32 tile to the output.
    const uint64_t out_gaddr = (uint64_t)(uintptr_t)(&out4[out_base]);
    const v4u sg0 = tdm_group0(lds_addr_of(&s_buf[0]), out_gaddr, /*gather=*/false);
    const v8i sg1 = tdm_group1(/*dim0*/D_N, /*dim1*/TILE_ROWS,
                               /*tile0*/D_N, /*tile1*/TILE_ROWS, /*tile2*/0,
                               /*stride0*/D_N);
    const v4i gz = (v4i){0, 0, 0, 0};
    __builtin_amdgcn_tensor_store_from_lds(sg0, sg1, gz, gz, g4z, /*cpol*/0);
    // S_ENDPGM's implicit wait-idle covers the outstanding TENSORcnt.
}

// ---------------------------------------------------------------------------
// Harness entry point.
//   d_in[0]: xs  (B,T,D) float32
//   d_in[1]: ds  (B,T)   integer (harness passes as int32)
//   d_out  : (B, MAX_FRAMES, D) float32
//   d_ws   : scratch; we use B*T ints (32 KB) for the cumsum
// ---------------------------------------------------------------------------
extern "C" void kernel_launch(void* const* d_in, const int* in_sizes, int n_in,
                              void* d_out, int out_size, void* d_ws, size_t ws_size,
                              hipStream_t stream) {
    const float* xs  = (const float*)d_in[0];
    const int*   ds  = (const int*)d_in[1];
    float*       out = (float*)d_out;
    int*         cs  = (int*)d_ws;

    lr_cumsum_kernel<<<B_N, 256, 0, stream>>>(ds, cs);

    dim3 grid(TILES_PB, B_N);            // 176 x 32 tiles, 1 wave each
    lr_tdm_kernel<<<grid, 32, 0, stream>>>(xs, cs, out);
}